// MultiEMAMixture_14396730377046
// MI455X (gfx1250) — compile-verified
//
#include <hip/hip_runtime.h>
#include <hip/hip_bf16.h>
#include <cstdint>
#include <cstddef>

// ---------------- problem constants (from reference) ----------------
#define B_ 32
#define T_ 4096
#define C_ 512
#define K_ 3
#define CLAMP_LO 1e-4f
#define CLAMP_HI (1.0f - 1e-4f)

// ---------------- chunked-scan configuration ----------------
#define LCH 128               // timesteps per chunk (power of two)
#define NCH (T_ / LCH)        // 32 chunks
#define TILE_T 16             // timesteps staged per LDS tile
#define NTILE (LCH / TILE_T)  // 8 tiles per chunk

static_assert((LCH & (LCH - 1)) == 0, "LCH must be a power of two");
static_assert(LCH * NCH == T_, "chunking must cover T");
static_assert(TILE_T * NTILE == LCH, "tiling must cover chunk");

// Workspace layout (floats):
//   [0,      K*C)              a[k][c]        per-channel decay
//   [K*C,    2K*C)             w[k][c]        mix weights
//   [2K*C,   3K*C)             a^LCH[k][c]
//   [3K*C,   3K*C + K*B*NCH*C) S (chunk summaries; rewritten in-place to Ystart)
#define WS_A 0
#define WS_W (K_ * C_)
#define WS_AP (2 * K_ * C_)
#define WS_S (3 * K_ * C_)
#define S_KSTRIDE ((size_t)B_ * NCH * C_)

// ---------------- CDNA5 async global->LDS path ----------------
typedef int v4i __attribute__((vector_size(16)));
typedef __attribute__((address_space(1))) v4i* gptr_v4i;
typedef __attribute__((address_space(3))) v4i* lptr_v4i;

#if __has_builtin(__builtin_amdgcn_global_load_async_to_lds_b128)
#define EMA_HAVE_ASYNC 1
#endif

__device__ __forceinline__ void ema_copy16_g2l(const float* __restrict__ g, float* l) {
#if defined(EMA_HAVE_ASYNC)
  // emits global_load_async_to_lds_b128 (tracked by ASYNCcnt)
  __builtin_amdgcn_global_load_async_to_lds_b128((gptr_v4i)g, (lptr_v4i)l, 0, 0);
#else
  // synchronous fallback: global load + ds store
  *reinterpret_cast<float4*>(l) = *reinterpret_cast<const float4*>(g);
#endif
}

__device__ __forceinline__ void ema_wait_async0() {
#if defined(EMA_HAVE_ASYNC)
#if __has_builtin(__builtin_amdgcn_s_wait_asynccnt)
  __builtin_amdgcn_s_wait_asynccnt(0);
#else
  asm volatile("s_wait_asynccnt 0" ::: "memory");
#endif
#endif
}

__device__ __forceinline__ void ema_wait_async4() {
#if defined(EMA_HAVE_ASYNC)
#if __has_builtin(__builtin_amdgcn_s_wait_asynccnt)
  __builtin_amdgcn_s_wait_asynccnt(4);
#else
  asm volatile("s_wait_asynccnt 4" ::: "memory");
#endif
#endif
}

// ---------------- phase 0: per-channel parameters ----------------
__global__ __launch_bounds__(256) void ema_params_kernel(
    const float* __restrict__ logit_alpha,  // [K, C]
    const float* __restrict__ mix_logits,   // [C, K]
    float* __restrict__ ws) {
  const int c = blockIdx.x * blockDim.x + threadIdx.x;
  if (c >= C_) return;

  float lg[K_];
#pragma unroll
  for (int k = 0; k < K_; ++k) lg[k] = mix_logits[c * K_ + k];
  float m = lg[0];
#pragma unroll
  for (int k = 1; k < K_; ++k) m = fmaxf(m, lg[k]);
  float e[K_];
  float s = 0.0f;
#pragma unroll
  for (int k = 0; k < K_; ++k) {
    e[k] = expf(lg[k] - m);
    s += e[k];
  }
  const float inv = 1.0f / s;

#pragma unroll
  for (int k = 0; k < K_; ++k) {
    float la = logit_alpha[k * C_ + c];
    float a = 1.0f / (1.0f + expf(-la));
    a = fminf(fmaxf(a, CLAMP_LO), CLAMP_HI);
    float ap = a;  // a^LCH via repeated squaring (LCH power of two)
    for (int t = LCH; t > 1; t >>= 1) ap *= ap;
    ws[WS_A + k * C_ + c] = a;
    ws[WS_W + k * C_ + c] = e[k] * inv;
    ws[WS_AP + k * C_ + c] = ap;
  }
}

// ---------------- phases 1 & 3: chunk-local scan ----------------
// WITH_OUT=false: compute chunk summary (chunk 0: true end state; others: zero-init
//                 local inhomogeneous sum) -> S.
// WITH_OUT=true : read Ystart from S (chunk>0), emit mixed outputs for every t.
// Each wave async-copies ONLY its own 32-channel stripe, so no barriers are needed,
// just per-wave s_wait_asynccnt (double-buffered TILE_T x 32 tiles).
template <bool WITH_OUT>
__global__ __launch_bounds__(256) void ema_scan_kernel(
    const float* __restrict__ x,  // [B, T, C]
    float* __restrict__ ws,
    float* __restrict__ out) {    // [B, T, C]
  __shared__ __align__(16) float smem[8][2][TILE_T][32];

  const int tid = threadIdx.x;
  const int lane = tid & 31;
  const int wv = tid >> 5;
  const int c = blockIdx.x * 256 + tid;           // this thread's channel
  const int c0w = blockIdx.x * 256 + (wv << 5);   // wave's channel stripe base
  const int ch = blockIdx.y;                      // chunk index
  const int b = blockIdx.z;                       // batch index

  const float a0 = ws[WS_A + 0 * C_ + c];
  const float a1 = ws[WS_A + 1 * C_ + c];
  const float a2 = ws[WS_A + 2 * C_ + c];
  float w0 = 0.f, w1 = 0.f, w2 = 0.f;
  if (WITH_OUT) {
    w0 = ws[WS_W + 0 * C_ + c];
    w1 = ws[WS_W + 1 * C_ + c];
    w2 = ws[WS_W + 2 * C_ + c];
  }

  float* S = ws + WS_S;
  const size_t sidx = (size_t)b * NCH * C_ + (size_t)ch * C_ + c;

  float y0, y1, y2;
  if (ch != 0 && WITH_OUT) {  // Ystart produced by the combine phase
    y0 = S[sidx];
    y1 = S[S_KSTRIDE + sidx];
    y2 = S[2 * S_KSTRIDE + sidx];
  } else {
    y0 = y1 = y2 = 0.0f;      // chunk 0 is seeded by x[t=0] below
  }

  // per-lane async copy source/dest bases (b128: lane moves 16B = 4 channels)
  const float* gx = x + ((size_t)b * T_ + (size_t)ch * LCH) * C_ + c0w;
  const int row0 = lane >> 3;          // 4 rows per b128 issue, 8 lanes per row
  const int col4 = (lane & 7) << 2;
  const float* gsrc = gx + (size_t)row0 * C_ + col4;
  float* ldst[2] = {&smem[wv][0][row0][col4], &smem[wv][1][row0][col4]};

  // prime the pipeline: tile 0 -> buffer 0
#pragma unroll
  for (int i = 0; i < 4; ++i) ema_copy16_g2l(gsrc + (size_t)(i * 4) * C_, ldst[0] + i * 4 * 32);

  float* outp = WITH_OUT ? out + ((size_t)b * T_ + (size_t)ch * LCH) * C_ + c : nullptr;

  int buf = 0;
#pragma unroll
  for (int tile = 0; tile < NTILE; ++tile) {
    if (tile + 1 < NTILE) {  // issue next tile into the other buffer, then wait current
      const float* gs2 = gsrc + (size_t)(tile + 1) * TILE_T * C_;
      float* ld2 = ldst[buf ^ 1];
#pragma unroll
      for (int i = 0; i < 4; ++i) ema_copy16_g2l(gs2 + (size_t)(i * 4) * C_, ld2 + i * 4 * 32);
      ema_wait_async4();
    } else {
      ema_wait_async0();
    }

    const float* tptr = &smem[wv][buf][0][lane];
#pragma unroll
    for (int tt = 0; tt < TILE_T; ++tt) {
      const float xv = tptr[tt * 32];
      if (ch == 0 && tile == 0 && tt == 0) {
        y0 = xv;  // reference: y_0 = x_0 for every EMA branch
        y1 = xv;
        y2 = xv;
      } else {
        // y = a*y + (1-a)*x  ==  fma(a, y-x, x)
        y0 = fmaf(a0, y0 - xv, xv);
        y1 = fmaf(a1, y1 - xv, xv);
        y2 = fmaf(a2, y2 - xv, xv);
      }
      if (WITH_OUT) {
        const float ov = fmaf(w2, y2, fmaf(w1, y1, w0 * y0));
        __builtin_nontemporal_store(ov, outp);
        outp += C_;
      }
    }
    buf ^= 1;
  }

  if (!WITH_OUT) {
    S[sidx] = y0;
    S[S_KSTRIDE + sidx] = y1;
    S[2 * S_KSTRIDE + sidx] = y2;
  }
}

// ---------------- phase 2: sequential chunk recombination ----------------
// In-place: S[k][b][ch][c] (local sums) -> Ystart for chunk ch.
// carry = true end-state of chunk 0; for ch>=1:
//   Ystart[ch] = carry; carry = a^L * carry + S_local[ch]
__global__ __launch_bounds__(256) void ema_combine_kernel(float* __restrict__ ws) {
  const int g = blockIdx.x * blockDim.x + threadIdx.x;  // 0 .. B*C-1
  const int b = g / C_;
  const int c = g % C_;
  float* S = ws + WS_S;
#pragma unroll
  for (int k = 0; k < K_; ++k) {
    const float aP = ws[WS_AP + k * C_ + c];
    const size_t base = (size_t)k * S_KSTRIDE + (size_t)b * NCH * C_ + c;
    float carry = S[base];  // true y at end of chunk 0
#pragma unroll 4
    for (int ch = 1; ch < NCH; ++ch) {
      const size_t idx = base + (size_t)ch * C_;
      const float tmp = S[idx];  // local sum of chunk ch
      S[idx] = carry;            // becomes Ystart for chunk ch
      carry = fmaf(aP, carry, tmp);
    }
  }
}

// ---------------- launch ----------------
extern "C" void kernel_launch(void* const* d_in, const int* in_sizes, int n_in,
                              void* d_out, int out_size, void* d_ws, size_t ws_size,
                              hipStream_t stream) {
  (void)in_sizes;
  (void)n_in;
  (void)out_size;
  (void)ws_size;
  const float* x = (const float*)d_in[0];            // [B, T, C]
  const float* logit_alpha = (const float*)d_in[1];  // [K, C]
  const float* mix_logits = (const float*)d_in[2];   // [C, K]
  float* out = (float*)d_out;                        // [B, T, C]
  float* ws = (float*)d_ws;                          // needs ~6.3 MB

  ema_params_kernel<<<dim3((C_ + 255) / 256), 256, 0, stream>>>(logit_alpha, mix_logits, ws);

  dim3 grid(C_ / 256, NCH, B_);  // (2, 32, 32) blocks of 256 threads
  ema_scan_kernel<false><<<grid, 256, 0, stream>>>(x, ws, nullptr);
  ema_combine_kernel<<<(B_ * C_) / 256, 256, 0, stream>>>(ws);
  ema_scan_kernel<true><<<grid, 256, 0, stream>>>(x, ws, out);
}